// LaplacianLoss_4776003633581
// MI455X (gfx1250) — compile-verified
//
#include <hip/hip_runtime.h>

typedef float v2f __attribute__((ext_vector_type(2)));
typedef float v8f __attribute__((ext_vector_type(8)));

constexpr int NVV            = 16384;
constexpr int ROWS_PER_BLOCK = 32;                    // two 16-row WMMA tiles per wave
constexpr int WAVES_PER_BLK  = 8;
constexpr int THREADS        = WAVES_PER_BLK * 32;    // 256 (wave32!)
constexpr int NUM_BLOCKS     = NVV / ROWS_PER_BLOCK;  // 512
constexpr int K_PER_WAVE     = NVV / WAVES_PER_BLK;   // 2048

// Phase 1: each block computes sum of squares of 32 rows of y = L @ x,
// K split across the 8 waves, partial D tiles reduced through LDS.
__global__ __launch_bounds__(THREADS) void lap_loss_partial(
    const float* __restrict__ x,   // (NVV, 3)
    const float* __restrict__ L,   // (NVV, NVV) row-major
    float* __restrict__ ws)        // (NUM_BLOCKS,) partial sums
{
    __shared__ float red[WAVES_PER_BLK * 512];  // 8 waves x (2 tiles x 8 vgpr x 32 lanes)
    __shared__ float rsum[THREADS];

    const int tid  = threadIdx.x;
    const int wave = tid >> 5;
    const int lane = tid & 31;
    const int m    = lane & 15;   // row-within-tile (A) / column N (B,D)
    const int hp   = lane >> 4;   // half selector

    const int  R      = blockIdx.x * ROWS_PER_BLOCK;
    const long kstart = (long)wave * K_PER_WAVE;
    const long kend   = kstart + K_PER_WAVE;

    // A layout (32-bit 16x4): lane m holds K = 2*hp, 2*hp+1 of row m -> float2 load
    const float* rowA0 = L + (long)(R + m)      * NVV + 2 * hp;
    const float* rowA1 = L + (long)(R + 16 + m) * NVV + 2 * hp;

    // B layout (4x16): v0 lane = row K=k0+hp col m ; v1 = row K=k0+2+hp col m.
    // Only columns 0..2 are real x data; mask the rest with *0 (branch-free:
    // EXEC stays all-1s as required by WMMA).
    const int   mn    = (m < 3) ? m : 2;
    const float bmask = (m < 3) ? 1.0f : 0.0f;

    v8f d0 = {0.f, 0.f, 0.f, 0.f, 0.f, 0.f, 0.f, 0.f};
    v8f d1 = {0.f, 0.f, 0.f, 0.f, 0.f, 0.f, 0.f, 0.f};

#pragma unroll 4
    for (long k0 = kstart; k0 < kend; k0 += 4) {
        v2f a0 = *(const v2f*)(rowA0 + k0);           // global_load_b64 (256 B/wave of L)
        v2f a1 = *(const v2f*)(rowA1 + k0);
        float b0 = x[(k0 + hp)     * 3 + mn] * bmask; // x is L2/WGP$ resident (192 KiB)
        float b1 = x[(k0 + 2 + hp) * 3 + mn] * bmask;
        v2f b = {b0, b1};
        // 8 args: (neg_a, A, neg_b, B, c_mod, C, reuse_a, reuse_b)
        d0 = __builtin_amdgcn_wmma_f32_16x16x4_f32(false, a0, false, b, (short)0, d0, false, false);
        d1 = __builtin_amdgcn_wmma_f32_16x16x4_f32(false, a1, false, b, (short)0, d1, false, false);
    }

    // Stash partial D tiles: layout across waves is identical, so summing
    // per-position then squaring is exact.
    float* myred = red + wave * 512;
#pragma unroll
    for (int r = 0; r < 8; ++r) {
        myred[r * 32 + lane]       = d0[r];
        myred[256 + r * 32 + lane] = d1[r];
    }
    __syncthreads();

    float sq = 0.0f;
#pragma unroll
    for (int p = tid; p < 512; p += THREADS) {
        float tot = 0.0f;
#pragma unroll
        for (int w = 0; w < WAVES_PER_BLK; ++w) tot += red[w * 512 + p];
        sq += tot * tot;   // columns 3..15 are exactly zero, harmless
    }
    rsum[tid] = sq;
    __syncthreads();
#pragma unroll
    for (int s = THREADS / 2; s > 0; s >>= 1) {
        if (tid < s) rsum[tid] += rsum[tid + s];
        __syncthreads();
    }
    if (tid == 0) ws[blockIdx.x] = rsum[0];
}

// Phase 2: deterministic tree reduction of the 512 per-block partials.
__global__ __launch_bounds__(NUM_BLOCKS) void lap_loss_final(
    const float* __restrict__ ws, float* __restrict__ out)
{
    __shared__ float s[NUM_BLOCKS];
    const int t = threadIdx.x;
    s[t] = ws[t];
    __syncthreads();
#pragma unroll
    for (int st = NUM_BLOCKS / 2; st > 0; st >>= 1) {
        if (t < st) s[t] += s[t + st];
        __syncthreads();
    }
    if (t == 0) out[0] = s[0];
}

extern "C" void kernel_launch(void* const* d_in, const int* in_sizes, int n_in,
                              void* d_out, int out_size, void* d_ws, size_t ws_size,
                              hipStream_t stream) {
    const float* x  = (const float*)d_in[0];  // (16384, 3)
    const float* L  = (const float*)d_in[1];  // (16384, 16384)
    float*       ws = (float*)d_ws;           // 512 floats of scratch
    (void)in_sizes; (void)n_in; (void)out_size; (void)ws_size;

    lap_loss_partial<<<NUM_BLOCKS, THREADS, 0, stream>>>(x, L, ws);
    lap_loss_final<<<1, NUM_BLOCKS, 0, stream>>>(ws, (float*)d_out);
}